// RGCN_42193758716296
// MI455X (gfx1250) — compile-verified
//
#include <hip/hip_runtime.h>

typedef __attribute__((ext_vector_type(2))) float v2f;
typedef __attribute__((ext_vector_type(8))) float v8f;

// N=100000, R=8, D=64, E=3200000
// agg layout per node: [r0..r7 sums (512 floats) | h (64 floats)] = 576 floats
#define KTOT 576

// Zero relation slots 0..7 of agg and the per-(node,rel) counters (float4-wide).
__global__ void rgcn_zero(float* __restrict__ agg, float* __restrict__ cnt, int N) {
    long long idx = (long long)blockIdx.x * blockDim.x + threadIdx.x;
    long long aggVecs = (long long)N * 128;            // N*512 floats
    float4 z = make_float4(0.f, 0.f, 0.f, 0.f);
    if (idx < aggVecs) {
        long long n = idx >> 7;
        int rem = (int)(idx & 127);
        ((float4*)(agg + n * 576))[rem] = z;
    } else {
        long long c = idx - aggVecs;
        if (c < (long long)N * 2) ((float4*)cnt)[c] = z;  // N*8 floats
    }
}

// h1[n,:] = emb[x[n],:]; also seed agg slot 8 with h for the fused GEMM.
__global__ void rgcn_gather(const int* __restrict__ x, const float* __restrict__ emb,
                            float* __restrict__ h, float* __restrict__ agg, int N) {
    long long idx = (long long)blockIdx.x * blockDim.x + threadIdx.x;
    long long n = idx >> 4;
    int q = (int)(idx & 15);
    if (n >= N) return;
    long long node = x[n];
    float4 v = ((const float4*)(emb + node * 64))[q];
    ((float4*)(h + n * 64))[q] = v;
    ((float4*)(agg + n * 576 + 512))[q] = v;
}

// 32 threads per edge, float2 per thread -> 64 f32 atomics/edge.
// Edge metadata loads are wave-uniform. Lane 0 bumps the (dst, rel) count.
__global__ void rgcn_scatter(const float* __restrict__ h, const int* __restrict__ src,
                             const int* __restrict__ dst, const int* __restrict__ et,
                             float* __restrict__ agg, float* __restrict__ cnt, long long E) {
    long long t = (long long)blockIdx.x * blockDim.x + threadIdx.x;
    long long e = t >> 5;
    int kp = (int)(t & 31);
    if (e >= E) return;
    long long s = src[e];
    long long d = dst[e];
    int r = et[e];
    float2 v = ((const float2*)(h + s * 64))[kp];
    float* base = agg + d * 576 + (long long)r * 64 + kp * 2;
    atomicAdd(base, v.x);
    atomicAdd(base + 1, v.y);
    if (kp == 0) atomicAdd(cnt + d * 8 + r, 1.0f);
}

// sums -> means: scale relation slots by 1/max(cnt,1) (float4-wide RMW).
__global__ void rgcn_normalize(float* __restrict__ agg, const float* __restrict__ cnt, int N) {
    long long idx = (long long)blockIdx.x * blockDim.x + threadIdx.x;
    if (idx >= (long long)N * 128) return;
    long long n = idx >> 7;
    int rem = (int)(idx & 127);
    int r = rem >> 4;
    float c = cnt[n * 8 + r];
    float s = 1.0f / fmaxf(c, 1.0f);
    float4* p = (float4*)(agg + n * 576) + rem;
    float4 v = *p;
    v.x *= s; v.y *= s; v.z *= s; v.w *= s;
    *p = v;
}

// Fused GEMM: out[n, 0:64] = A[n, 0:576] @ [W(512x64); root(64x64)] + bias.
// The stacked B matrix (147456 B) is staged once per workgroup into LDS
// (320 KB WGP pool) in k-pair-interleaved layout: element (k, c) lives at
// Bs[(k>>1)*128 + c*2 + (k&1)], so each WMMA B fragment is a single aligned
// ds_load_b64 straight into an even VGPR pair (no repacking moves). Each wave
// computes a 16x64 strip with 4 accumulators via V_WMMA_F32_16X16X4_F32,
// reusing the A fragment across the 4 column tiles. Optionally fuses ReLU and
// mirrors the output into agg slot 8 for the next layer's A matrix.
__global__ void __launch_bounds__(256)
rgcn_gemm(const float* __restrict__ A, const float* __restrict__ W,
          const float* __restrict__ root, const float* __restrict__ bias,
          float* __restrict__ out, float* __restrict__ aggs8,
          int relu, int N) {
    __shared__ __align__(16) float Bs[KTOT * 64];   // [W(512x64); root(64x64)], pair-interleaved

    // Cooperative stage of B into LDS with the k-pair interleave.
    for (int i = threadIdx.x; i < 288 * 64; i += 256) {   // 288 k-pairs x 64 cols
        int p = i >> 6;          // k-pair index
        int c = i & 63;          // column
        int k = p << 1;
        float v0, v1;
        if (k < 512) {
            v0 = W[k * 64 + c];
            v1 = W[(k + 1) * 64 + c];
        } else {
            v0 = root[(k - 512) * 64 + c];
            v1 = root[(k - 511) * 64 + c];
        }
        Bs[p * 128 + c * 2]     = v0;
        Bs[p * 128 + c * 2 + 1] = v1;
    }
    __syncthreads();

    int wave = (int)(((long long)blockIdx.x * blockDim.x + threadIdx.x) >> 5);
    int lane = threadIdx.x & 31;
    int nTiles = N >> 4;                        // N = 100000 -> 6250 row tiles
    if (wave >= nTiles) return;
    long long m0 = (long long)wave * 16;
    int mlane = lane & 15;                      // row (A) / column (B,C,D) in tile
    int grp = lane >> 4;                        // lane half: K pair / output row half

    const float* Arow = A + (m0 + mlane) * 576;
    v8f acc[4] = {};

    for (int k0 = 0; k0 < KTOT; k0 += 4) {
        // A fragment (16x4 f32): lane half g holds K = k0+2g, k0+2g+1 for row mlane
        int ka = k0 + grp * 2;                  // even -> 8B-aligned global_load_b64
        v2f a = *(const v2f*)(Arow + ka);
        int p = ka >> 1;                        // k-pair row in Bs
#pragma unroll
        for (int nt = 0; nt < 4; ++nt) {
            int c = nt * 16 + mlane;
            v2f b = *(const v2f*)(Bs + p * 128 + c * 2);   // aligned ds_load_b64
            acc[nt] = __builtin_amdgcn_wmma_f32_16x16x4_f32(
                false, a, false, b, (short)0, acc[nt], false, false);
        }
    }

#pragma unroll
    for (int nt = 0; nt < 4; ++nt) {
#pragma unroll
        for (int i = 0; i < 8; ++i) {
            long long m = m0 + grp * 8 + i;     // C/D layout: VGPR i -> row grp*8+i
            int c = nt * 16 + mlane;
            float v = acc[nt][i] + bias[c];
            if (relu) v = fmaxf(v, 0.0f);
            out[m * 64 + c] = v;
            if (aggs8) aggs8[m * 576 + 512 + c] = v;
        }
    }
}

extern "C" void kernel_launch(void* const* d_in, const int* in_sizes, int n_in,
                              void* d_out, int out_size, void* d_ws, size_t ws_size,
                              hipStream_t stream) {
    const int*   x     = (const int*)d_in[0];
    const int*   eidx  = (const int*)d_in[1];
    const int*   etype = (const int*)d_in[2];
    const float* emb   = (const float*)d_in[3];
    const float* W1    = (const float*)d_in[4];
    const float* root1 = (const float*)d_in[5];
    const float* b1    = (const float*)d_in[6];
    const float* W2    = (const float*)d_in[7];
    const float* root2 = (const float*)d_in[8];
    const float* b2    = (const float*)d_in[9];
    float* outp = (float*)d_out;

    int N = in_sizes[0];
    long long E = (long long)in_sizes[1] / 2;
    const int* src = eidx;
    const int* dst = eidx + E;

    // Workspace carving (floats): agg[N*576] | h1[N*64] | h2[N*64] | cnt[N*8]
    float* agg = (float*)d_ws;
    float* h1  = agg + (size_t)N * 576;
    float* h2  = h1 + (size_t)N * 64;
    float* cnt = h2 + (size_t)N * 64;

    const int B = 256;
    long long zThreads = (long long)N * 128 + (long long)N * 2;   // float4 units
    int zBlocks = (int)((zThreads + B - 1) / B);
    long long gThreads = (long long)N * 16;                       // float4 units
    int gBlocks = (int)((gThreads + B - 1) / B);
    long long sThreads = E * 32;                                  // float2 units
    int sBlocks = (int)((sThreads + B - 1) / B);
    long long nThreads = (long long)N * 128;                      // float4 units
    int nBlocks = (int)((nThreads + B - 1) / B);
    int waves = (N + 15) / 16;
    int mBlocks = (waves * 32 + B - 1) / B;

    // ---- Layer 1 ----
    rgcn_zero<<<zBlocks, B, 0, stream>>>(agg, cnt, N);
    rgcn_gather<<<gBlocks, B, 0, stream>>>(x, emb, h1, agg, N);
    rgcn_scatter<<<sBlocks, B, 0, stream>>>(h1, src, dst, etype, agg, cnt, E);
    rgcn_normalize<<<nBlocks, B, 0, stream>>>(agg, cnt, N);
    // writes ReLU(out) into h2 and into agg slot 8 (A matrix for layer 2)
    rgcn_gemm<<<mBlocks, B, 0, stream>>>(agg, W1, root1, b1, h2, agg, 1, N);

    // ---- Layer 2 ----
    rgcn_zero<<<zBlocks, B, 0, stream>>>(agg, cnt, N);   // slot 8 (h2 copy) preserved
    rgcn_scatter<<<sBlocks, B, 0, stream>>>(h2, src, dst, etype, agg, cnt, E);
    rgcn_normalize<<<nBlocks, B, 0, stream>>>(agg, cnt, N);
    rgcn_gemm<<<mBlocks, B, 0, stream>>>(agg, W2, root2, b2, outp, nullptr, 0, N);
}